// LSTM_5291399709001
// MI455X (gfx1250) — compile-verified
//
#include <hip/hip_runtime.h>
#include <hip/hip_bf16.h>

typedef __attribute__((ext_vector_type(2))) float v2f;
typedef __attribute__((ext_vector_type(8))) float v8f;

// Problem dims (match reference)
constexpr int Bb = 32;
constexpr int Tt = 2048;
constexpr int Dd = 512;
constexpr int Hh = 512;
constexpr int G4 = 4 * Hh;      // 2048 gate columns

// Persistent-kernel geometry: 256 gate tiles (2 x 128), one wave32 per tile.
constexpr int NWG  = 32;        // workgroups (all co-resident)
constexpr int NTHR = 256;       // 8 wave32s per workgroup

__device__ __forceinline__ float sigmoidf_(float v) {
    return 1.0f / (1.0f + __expf(-v));
}

// ---------------------------------------------------------------------------
// Init kernel: copy h0/c0 into scratch state, zero barrier counters.
// ---------------------------------------------------------------------------
__global__ void lstm_init(const float* __restrict__ h0, const float* __restrict__ c0,
                          float* __restrict__ hbuf, float* __restrict__ cbuf,
                          unsigned* __restrict__ ctrs) {
    int i = blockIdx.x * blockDim.x + threadIdx.x;
    for (int e = i; e < Bb * Hh; e += gridDim.x * blockDim.x) {
        hbuf[e] = h0[e];
        cbuf[e] = c0[e];
    }
    if (i == 0) { ctrs[0] = 0u; ctrs[1] = 0u; }
}

// ---------------------------------------------------------------------------
// Device-wide barrier (sense via generation counter). All NWG workgroups are
// resident, so spin-wait is safe. Fences give acquire/release semantics so
// gates/h/c written before the barrier are visible after it.
// ---------------------------------------------------------------------------
__device__ __forceinline__ void grid_sync(unsigned* cnt, unsigned* gen) {
    __syncthreads();
    if (threadIdx.x == 0) {
        __threadfence();
        unsigned g = __hip_atomic_load(gen, __ATOMIC_RELAXED, __HIP_MEMORY_SCOPE_AGENT);
        unsigned arrived =
            __hip_atomic_fetch_add(cnt, 1u, __ATOMIC_ACQ_REL, __HIP_MEMORY_SCOPE_AGENT);
        if (arrived == NWG - 1) {
            __hip_atomic_store(cnt, 0u, __ATOMIC_RELAXED, __HIP_MEMORY_SCOPE_AGENT);
            __hip_atomic_fetch_add(gen, 1u, __ATOMIC_RELEASE, __HIP_MEMORY_SCOPE_AGENT);
        } else {
            while (__hip_atomic_load(gen, __ATOMIC_ACQUIRE, __HIP_MEMORY_SCOPE_AGENT) == g) {
                __builtin_amdgcn_s_sleep(1);
            }
        }
        __threadfence();
    }
    __syncthreads();
}

// ---------------------------------------------------------------------------
// Persistent LSTM kernel. Each wave owns one 16x16 tile of the 32x2048 gate
// matrix for all 2048 timesteps:
//   gates = [x_t ; h_{t-1}] (16 x 1024) @ W (1024 x 16), via fp32 WMMA K=4.
// Then a grid barrier, elementwise cell update (2 elems/thread), barrier.
// ---------------------------------------------------------------------------
__global__ __launch_bounds__(NTHR, 1)
void lstm_persistent(const float* __restrict__ x, const float* __restrict__ W,
                     const float* __restrict__ bias, float* __restrict__ out,
                     float* __restrict__ gates, float* __restrict__ hbuf,
                     float* __restrict__ cbuf, unsigned* __restrict__ ctrs) {
    constexpr int PADH = Dd + 4;            // 516: stride ≡ 4 mod 64 banks -> conflict-free
    __shared__ float hsh[16 * PADH];        // this workgroup's 16 rows of h (33 KB)

    const int wg    = blockIdx.x;           // 0..31
    const int tid   = threadIdx.x;
    const int lane  = tid & 31;
    const int wv    = tid >> 5;             // wave in workgroup, 0..7
    const int mTile = wg >> 4;              // wg 0..15 -> batch rows 0..15; 16..31 -> 16..31
    const int m0    = mTile * 16;
    const int nTile = (wg * 8 + wv) & 127;  // 0..127
    const int n0    = nTile * 16;

    const int l16   = lane & 15;            // M (for A) / N (for B,C)
    const int khalf = (lane >> 4) * 2;      // K-half select for A/B fragments
    const int arow  = m0 + l16;             // batch row this lane's A fragment covers

    unsigned* cnt = ctrs;
    unsigned* gen = ctrs + 1;

    const float* Wh = W + (size_t)Dd * G4;  // rows D..D+H of W

    for (int t = 0; t < Tt; ++t) {
        // Stage the 16 h rows this workgroup needs into LDS (no divergence:
        // 8192 elements / 256 threads = exactly 32 iterations each).
        for (int idx = tid; idx < 16 * Hh; idx += NTHR) {
            int r = idx >> 9;               // / 512
            int c = idx & (Hh - 1);
            hsh[r * PADH + c] = hbuf[(m0 + r) * Hh + c];
        }
        __syncthreads();

        v8f acc = {};

        // ---- x_t @ Wx contribution (K = 512) ----
        const float* xrow = x + (size_t)arow * Tt * Dd + (size_t)t * Dd;
#pragma unroll 4
        for (int k = 0; k < Dd; k += 4) {
            int ka = k + khalf;
            v2f a;  a.x = xrow[ka];
                    a.y = xrow[ka + 1];
            v2f bf; bf.x = W[(size_t)ka       * G4 + n0 + l16];
                    bf.y = W[(size_t)(ka + 1) * G4 + n0 + l16];
            acc = __builtin_amdgcn_wmma_f32_16x16x4_f32(false, a, false, bf,
                                                        (short)0, acc, false, false);
        }

        // ---- h_{t-1} @ Wh contribution (K = 512), A from LDS ----
#pragma unroll 4
        for (int k = 0; k < Hh; k += 4) {
            int ka = k + khalf;
            v2f a;  a.x = hsh[l16 * PADH + ka];
                    a.y = hsh[l16 * PADH + ka + 1];
            v2f bf; bf.x = Wh[(size_t)ka       * G4 + n0 + l16];
                    bf.y = Wh[(size_t)(ka + 1) * G4 + n0 + l16];
            acc = __builtin_amdgcn_wmma_f32_16x16x4_f32(false, a, false, bf,
                                                        (short)0, acc, false, false);
        }

        // Store gate tile: VGPR r holds M = r + 8*lane[4], N = n0 + lane[3:0].
#pragma unroll
        for (int r = 0; r < 8; ++r) {
            int row = m0 + r + ((lane >> 4) << 3);
            gates[row * G4 + n0 + l16] = acc[r];
        }

        grid_sync(cnt, gen);   // all gate tiles visible

        // Elementwise cell update: 16384 elems over 8192 threads -> 2 each.
        const int gtid = wg * NTHR + tid;
#pragma unroll
        for (int e = gtid; e < Bb * Hh; e += NWG * NTHR) {
            int bi = e >> 9;               // batch
            int hh = e & (Hh - 1);         // hidden index
            const float* grow = gates + bi * G4;
            float ig = sigmoidf_(grow[hh]           + bias[hh]);
            float fg = sigmoidf_(grow[Hh + hh]      + bias[Hh + hh]);
            float gg = tanhf    (grow[2 * Hh + hh]  + bias[2 * Hh + hh]);
            float og = sigmoidf_(grow[3 * Hh + hh]  + bias[3 * Hh + hh]);
            float c  = fg * cbuf[e] + ig * gg;
            float h  = og * tanhf(c);
            cbuf[e] = c;
            hbuf[e] = h;
            out[(size_t)bi * Tt * Hh + (size_t)t * Hh + hh] = h;
        }

        grid_sync(cnt, gen);   // h/c updated and gates consumed before next step
    }

    // Final (h_T, c_T) appended after outputs.
    float* outh = out + (size_t)Bb * Tt * Hh;
    float* outc = outh + Bb * Hh;
    for (int e = wg * NTHR + tid; e < Bb * Hh; e += NWG * NTHR) {
        outh[e] = hbuf[e];
        outc[e] = cbuf[e];
    }
}

// ---------------------------------------------------------------------------
// Launch: init state + counters, then one persistent kernel for all steps.
// ws layout (floats): gates[32*2048] | hbuf[32*512] | cbuf[32*512] | 2 u32 ctrs
// ---------------------------------------------------------------------------
extern "C" void kernel_launch(void* const* d_in, const int* in_sizes, int n_in,
                              void* d_out, int out_size, void* d_ws, size_t ws_size,
                              hipStream_t stream) {
    const float* x    = (const float*)d_in[0];
    const float* W    = (const float*)d_in[1];
    const float* bias = (const float*)d_in[2];
    const float* h0   = (const float*)d_in[3];
    const float* c0   = (const float*)d_in[4];
    float* out = (float*)d_out;

    float* gates = (float*)d_ws;
    float* hbuf  = gates + (size_t)Bb * G4;
    float* cbuf  = hbuf + (size_t)Bb * Hh;
    unsigned* ctrs = (unsigned*)(cbuf + (size_t)Bb * Hh);

    lstm_init<<<64, 256, 0, stream>>>(h0, c0, hbuf, cbuf, ctrs);
    lstm_persistent<<<NWG, NTHR, 0, stream>>>(x, W, bias, out, gates, hbuf, cbuf, ctrs);
}